// Graph_sampler_31628139168232
// MI455X (gfx1250) — compile-verified
//
#include <hip/hip_runtime.h>
#include <math.h>

typedef __attribute__((ext_vector_type(2))) float v2f;
typedef __attribute__((ext_vector_type(8))) float v8f;

#define NB 256      // batch
#define NC 3
#define NH 224
#define NW 224
#define NS 20       // samples per image
#define NM 10       // mixture components
#define NFEAT 5     // C + 2
#define NHID 100
#define NOUT 10
#define KP1 8       // GEMM1 K padded (5 -> 8)
#define CT1 7       // GEMM1 column tiles (100 -> 112)

// Single workgroup, 256 threads (8 wave32). All intermediates in LDS.
__global__ __launch_bounds__(256) void graph_sampler_kernel(
    const float* __restrict__ x,      // (256,3,224,224)
    const float* __restrict__ u_cat,  // (20,256)
    const float* __restrict__ z,      // (20,256,2)
    const float* __restrict__ mgp,    // (1,60)
    const float* __restrict__ w1,     // (5,100)
    const float* __restrict__ b1,     // (100)
    const float* __restrict__ w2,     // (100,10)
    const float* __restrict__ b2v,    // (10)
    const float* __restrict__ wfc,    // (200,10)
    const float* __restrict__ bf,     // (10)
    float* __restrict__ out)          // (256*10) out ++ (256) full_log_prob
{
  __shared__ float s_big[NS * NB * NFEAT];   // 25600 f: samples, then reused as h[256][100]
  __shared__ float s_gpre[NB * NOUT];        // pre-log-softmax g
  __shared__ float s_fsum[NB * KP1];         // A matrix for GEMM1 (K zero-padded)
  __shared__ float s_param[6 * NM];
  __shared__ float s_w1t[CT1 * 16 * KP1];    // w1 transposed+padded: [col][k], 112x8
  __shared__ float s_w2t[16 * NHID];         // w2 transposed+padded: [n][k], 16x100
  __shared__ float s_b1p[CT1 * 16];          // b1 padded to 112
  __shared__ float s_b2p[16];                // b2 padded to 16
  __shared__ float s_wf[NOUT * NOUT];        // w_fcf collapsed over s
  __shared__ float s_lse[NOUT];

  const int t = threadIdx.x;
  const int b = t;  // one thread per batch item

  // ---- stage small tensors into LDS (single pass per location: no races) ----
  if (t < 6 * NM) s_param[t] = mgp[t];
  for (int i = t; i < CT1 * 16 * KP1; i += 256) {       // w1t[col*8+k]
    int col = i / KP1, k = i % KP1;
    s_w1t[i] = (col < NHID && k < NFEAT) ? w1[k * NHID + col] : 0.f;
  }
  for (int i = t; i < 16 * NHID; i += 256) {            // w2t[n*100+k]
    int n = i / NHID, k = i % NHID;
    s_w2t[i] = (n < NOUT) ? w2[k * NOUT + n] : 0.f;
  }
  if (t < CT1 * 16) s_b1p[t] = (t < NHID) ? b1[t] : 0.f;
  if (t < 16)       s_b2p[t] = (t < NOUT) ? b2v[t] : 0.f;
  if (t < NOUT * NOUT) {
    // wf[o][o2] = sum_s w_fcf[(s*10+o)*10 + o2]  (g is s-independent)
    int o = t / NOUT, o2 = t % NOUT;
    float acc = 0.f;
    for (int s = 0; s < NS; ++s) acc += wfc[s * NOUT * NOUT + o * NOUT + o2];
    s_wf[t] = acc;
  }
  __syncthreads();

  // ---- Phase A: mixture CDF per b, then per-s sampling / gather ----
  // P(c,b,m) = param[((b*M+m)/B)*6 + c]   (broadcast+reshape collapses to this)
  float cdf[NM];
  {
    float p0[NM];
    float mx = -1e30f;
#pragma unroll
    for (int m = 0; m < NM; ++m) {
      int base = ((b * NM + m) >> 8) * 6;   // B == 256
      p0[m] = s_param[base];
      mx = fmaxf(mx, p0[m]);
    }
    float sum = 0.f;
#pragma unroll
    for (int m = 0; m < NM; ++m) { p0[m] = expf(p0[m] - mx); sum += p0[m]; }
    float run = 0.f, inv = 1.f / sum;
#pragma unroll
    for (int m = 0; m < NM; ++m) { run += p0[m]; cdf[m] = run * inv; }
  }

  float flp = 0.f;
  const float TWO_PI = 6.283185307179586f;
  for (int s = 0; s < NS; ++s) {
    float u = u_cat[s * NB + b];
    int idx = 0; bool found = false;
#pragma unroll
    for (int m = 0; m < NM; ++m) {          // argmax(u < cdf) = first true, else 0
      if (!found && u < cdf[m]) { idx = m; found = true; }
    }
    int base = ((b * NM + idx) >> 8) * 6;
    float mux = s_param[base + 1];
    float muy = s_param[base + 2];
    float sx  = expf(s_param[base + 3]);
    float sy  = expf(s_param[base + 4]);
    float r   = tanhf(s_param[base + 5]);
    float z1  = z[(s * NB + b) * 2 + 0];
    float z2  = z[(s * NB + b) * 2 + 1];
    float omr2 = 1.f - r * r;
    float sq   = sqrtf(omr2);
    float px = mux + sx * z1;
    float py = muy + sy * (r * z1 + sq * z2);
    float zx = (px - mux) / sx;
    float zy = (py - muy) / sy;
    float lp = -(zx * zx - 2.f * r * zx * zy + zy * zy) / (2.f * omr2)
               - logf(TWO_PI * sx * sy * sq);
    flp += lp;
    float lx = tanhf(px), ly = tanhf(py);
    int ri = (int)truncf(0.5f * (lx + 1.f) * (float)NH - 0.1f);
    ri = min(max(ri, 0), NH - 1);
    int ci = (int)truncf(0.5f * (ly + 1.f) * (float)NW - 0.1f);
    ci = min(max(ci, 0), NW - 1);
    const float* xp = x + ((size_t)b * NC) * NH * NW + (size_t)ri * NW + ci;
    int row = s * NB + b;                   // samples row (s-major) == feat row b*20+s
    s_big[row * NFEAT + 0] = xp[0];
    s_big[row * NFEAT + 1] = xp[NH * NW];
    s_big[row * NFEAT + 2] = xp[2 * NH * NW];
    s_big[row * NFEAT + 3] = lx;
    s_big[row * NFEAT + 4] = ly;
  }
  __syncthreads();

  // ---- Phase B1: adj collapse -> fsum, staged as zero-padded A matrix ----
  {
    float fsum[NFEAT] = {0.f, 0.f, 0.f, 0.f, 0.f};
    for (int s2 = 0; s2 < NS; ++s2) {
      int row = b * NS + s2;                // feat[b, s2, :] = samples_row[b*20+s2]
#pragma unroll
      for (int f = 0; f < NFEAT; ++f) fsum[f] += s_big[row * NFEAT + f];
    }
#pragma unroll
    for (int f = 0; f < NFEAT; ++f) s_fsum[b * KP1 + f] = fsum[f];
#pragma unroll
    for (int f = NFEAT; f < KP1; ++f) s_fsum[b * KP1 + f] = 0.f;
  }
  __syncthreads();                          // samples consumed; s_big becomes h

  const int lane = t & 31;
  const int wave = t >> 5;
  const int half = lane >> 4;               // 0: lanes 0-15, 1: lanes 16-31
  const int nloc = lane & 15;

  // ---- GEMM1 via WMMA: h = relu(fsum(256x8) @ w1t^T(8x112) + b1) ----
  // 16 row tiles x 7 col tiles = 112 jobs; 8 waves -> 14 jobs each; K = 8 -> 2 steps.
  for (int job = wave; job < 16 * CT1; job += 8) {
    const int tile = job / CT1;             // row tile of b
    const int ct   = job % CT1;             // col tile of hidden
    const int rowA = tile * 16 + nloc;
    const int col  = ct * 16 + nloc;
    v8f acc = {0.f, 0.f, 0.f, 0.f, 0.f, 0.f, 0.f, 0.f};
#pragma unroll
    for (int kk = 0; kk < KP1; kk += 4) {
      const int ka = kk + (half ? 2 : 0);   // A/B VGPR0: K{0,2}, VGPR1: K{1,3}
      v2f av, bv;
      av.x = s_fsum[rowA * KP1 + ka];
      av.y = s_fsum[rowA * KP1 + ka + 1];
      bv.x = s_w1t[col * KP1 + ka];
      bv.y = s_w1t[col * KP1 + ka + 1];
      acc = __builtin_amdgcn_wmma_f32_16x16x4_f32(
          false, av, false, bv, (short)0, acc, false, false);
    }
    const float bias = s_b1p[col];
    if (col < NHID) {
#pragma unroll
      for (int i = 0; i < 8; ++i) {         // D: VGPR i -> row (i + half*8)
        int mrow = tile * 16 + i + half * 8;
        s_big[mrow * NHID + col] = fmaxf(acc[i] + bias, 0.f);   // h[mrow][col]
      }
    }
  }
  __syncthreads();

  // ---- GEMM2 via WMMA: g = 20 * (h(256x100) @ w2(100x10pad16)) + b2 ----
  // 16 row tiles; wave w -> tiles w, w+8; K = 100 -> 25 steps.
  {
    const float bias_o = s_b2p[nloc];
    for (int tile = wave; tile < NB / 16; tile += 8) {
      v8f acc = {0.f, 0.f, 0.f, 0.f, 0.f, 0.f, 0.f, 0.f};
      const int rowA = tile * 16 + nloc;
      for (int kk = 0; kk < NHID; kk += 4) {
        const int ka = kk + (half ? 2 : 0);
        v2f av, bv;
        av.x = s_big[rowA * NHID + ka];     // contiguous pair -> ds_load_b64
        av.y = s_big[rowA * NHID + ka + 1];
        bv.x = s_w2t[nloc * NHID + ka];     // transposed+padded -> ds_load_b64
        bv.y = s_w2t[nloc * NHID + ka + 1];
        acc = __builtin_amdgcn_wmma_f32_16x16x4_f32(
            false, av, false, bv, (short)0, acc, false, false);
      }
      if (nloc < NOUT) {
#pragma unroll
        for (int i = 0; i < 8; ++i) {
          int mrow = tile * 16 + i + half * 8;
          s_gpre[mrow * NOUT + nloc] = 20.0f * acc[i] + bias_o;  // adj sum = *S
        }
      }
    }
  }
  __syncthreads();

  // ---- Phase C: log-softmax over batch axis (per output o) ----
  if (t < NOUT) {
    float mx = -1e30f;
    for (int bb = 0; bb < NB; ++bb) mx = fmaxf(mx, s_gpre[bb * NOUT + t]);
    float sm = 0.f;
    for (int bb = 0; bb < NB; ++bb) sm += expf(s_gpre[bb * NOUT + t] - mx);
    s_lse[t] = mx + logf(sm);
  }
  __syncthreads();

  // ---- Phase D: final 10x10 FC + softmax, write outputs ----
  {
    float gl[NOUT], lg[NOUT];
#pragma unroll
    for (int o = 0; o < NOUT; ++o) gl[o] = s_gpre[b * NOUT + o] - s_lse[o];
    float mx = -1e30f;
#pragma unroll
    for (int o2 = 0; o2 < NOUT; ++o2) {
      float a = bf[o2];
#pragma unroll
      for (int o = 0; o < NOUT; ++o) a += gl[o] * s_wf[o * NOUT + o2];
      lg[o2] = a;
      mx = fmaxf(mx, a);
    }
    float sm = 0.f;
#pragma unroll
    for (int o2 = 0; o2 < NOUT; ++o2) { lg[o2] = expf(lg[o2] - mx); sm += lg[o2]; }
    float inv = 1.f / sm;
#pragma unroll
    for (int o2 = 0; o2 < NOUT; ++o2) out[b * NOUT + o2] = lg[o2] * inv;
    out[NB * NOUT + b] = flp;               // full_log_prob appended
  }
}

extern "C" void kernel_launch(void* const* d_in, const int* in_sizes, int n_in,
                              void* d_out, int out_size, void* d_ws, size_t ws_size,
                              hipStream_t stream) {
  (void)in_sizes; (void)n_in; (void)out_size; (void)d_ws; (void)ws_size;
  graph_sampler_kernel<<<1, 256, 0, stream>>>(
      (const float*)d_in[0],  // x
      (const float*)d_in[1],  // u_cat
      (const float*)d_in[2],  // z
      (const float*)d_in[3],  // m_g_params
      (const float*)d_in[4],  // w_gc1
      (const float*)d_in[5],  // b_gc1
      (const float*)d_in[6],  // w_gc2
      (const float*)d_in[7],  // b_gc2
      (const float*)d_in[8],  // w_fcf
      (const float*)d_in[9],  // b_fcf
      (float*)d_out);
}